// MySubClassModel_55490977464599
// MI455X (gfx1250) — compile-verified
//
#include <hip/hip_runtime.h>
#include <stdint.h>

// Problem constants (fixed by setup_inputs: B=128, H=W=28, C=512, n=28).
#define HW    784          // H*W
#define NROW  28
#define NCH   512          // C
#define TP1   29           // n+1 (template table edge)

constexpr int CPB   = 128; // channels per block == threads per block (4 waves)
constexpr int TILE  = 28;  // spatial rows staged per async tile
constexpr int NTLS  = HW / TILE; // 28 tiles cover all 784 positions

// Fused peak-argmax + template-gather + relu-mask kernel.
// out layout: [masked (S)] [x copy (S)] [templates (S)], S = B*HW*C.
__global__ void peak_mask_kernel(const float* __restrict__ x,
                                 const float* __restrict__ tp,
                                 float* __restrict__ out,
                                 size_t S) {
    __shared__ float smem[2 * TILE * CPB];   // double buffer: 2 x 14 KB

    const int tid = threadIdx.x;
    const int b   = blockIdx.x / (NCH / CPB);
    const int c0  = (blockIdx.x % (NCH / CPB)) * CPB;
    const int c   = c0 + tid;
    const size_t xbase = (size_t)b * HW * NCH;

    // Async-copy lane mapping: one wave loads one whole row of the tile.
    // chunk in [0,32): which 16B (4-channel) slot; rowg in [0,4): wave id.
    const int chunk = tid & (CPB / 4 - 1);
    const int rowg  = tid / (CPB / 4);
    // Low 32 bits of the shared-aperture flat address == LDS byte offset.
    const unsigned lds0 = (unsigned)(size_t)(void*)smem;

    // Issue one tile (28 rows x 128 ch) of async global->LDS B128 copies.
    auto issue_tile = [&](int t) {
        const int pbase      = t * TILE;
        const unsigned bufo  = lds0 + (unsigned)((t & 1) * TILE * CPB * 4);
#pragma unroll
        for (int r = 0; r < TILE / 4; ++r) {
            const int row = r * 4 + rowg;
            const unsigned lofs = bufo + (unsigned)((row * CPB + chunk * 4) * 4);
            const unsigned long long ga =
                (unsigned long long)(size_t)(const void*)
                (x + xbase + (size_t)(pbase + row) * NCH + (size_t)(c0 + chunk * 4));
            asm volatile("global_load_async_to_lds_b128 %0, %1, off"
                         :: "v"(lofs), "v"(ga)
                         : "memory");
        }
    };

    // ---- Pass 1: argmax over the 784 spatial positions for channel c ----
    float mx  = -__builtin_inff();
    int   arg = 0;

    issue_tile(0);
    for (int t = 0; t < NTLS; ++t) {
        if (t + 1 < NTLS) {
            issue_tile(t + 1);                     // prefetch next tile
            asm volatile("s_wait_asynccnt 7" ::: "memory");  // tile t done
        } else {
            asm volatile("s_wait_asynccnt 0" ::: "memory");
        }
        __syncthreads();   // all 4 waves' async data visible in LDS

        const float* buf = &smem[(t & 1) * TILE * CPB];
#pragma unroll
        for (int r = 0; r < TILE; ++r) {
            const float v = buf[r * CPB + tid];    // conflict-free: stride 1
            const int   p = t * TILE + r;
            if (v > mx) { mx = v; arg = p; }       // strict > => first occurrence
        }
        __syncthreads();   // done reading buf before it is refilled at t+2
    }

    // ---- Pass 2: gather template, relu-mask, emit three outputs ----
    const int pi = arg / NROW;
    const int pj = arg % NROW;
    const size_t tbase = (size_t)(pi * TP1 + pj) * HW;  // template is [28,28] row-major

#pragma unroll 4
    for (int p = 0; p < HW; ++p) {
        const size_t idx = xbase + (size_t)p * NCH + (size_t)c;
        const float xv = x[idx];            // coalesced; L2-resident (NT stores
                                            // below keep x from being evicted)
        const float tv = tp[tbase + p];     // per-thread sequential template walk
        float m = xv * tv;
        m = m > 0.0f ? m : 0.0f;
        // Outputs are write-once / never re-read: non-temporal stores keep the
        // 616 MB of results from churning the 192 MB L2.
        __builtin_nontemporal_store(m,  &out[idx]);          // relu(x * templates)
        __builtin_nontemporal_store(xv, &out[S + idx]);      // x passthrough
        __builtin_nontemporal_store(tv, &out[2 * S + idx]);  // templates [B,n,n,C]
    }
}

extern "C" void kernel_launch(void* const* d_in, const int* in_sizes, int n_in,
                              void* d_out, int out_size, void* d_ws, size_t ws_size,
                              hipStream_t stream) {
    const float* x  = (const float*)d_in[0];
    const float* tp = (const float*)d_in[1];
    float* out      = (float*)d_out;

    const size_t S = (size_t)in_sizes[0];        // B*HW*C
    const int B    = (int)(S / ((size_t)HW * NCH));
    dim3 grid((unsigned)(B * (NCH / CPB)));      // 512 blocks
    dim3 block(CPB);                             // 128 threads = 4 waves

    peak_mask_kernel<<<grid, block, 0, stream>>>(x, tp, out, S);
}